// LowToHighMultiLevelReconstruction_87883620810891
// MI455X (gfx1250) — compile-verified
//
#include <hip/hip_runtime.h>
#include <math.h>

#define B_   2
#define N_   8192
#define DIM_ 512
#define GS_  128
#define WS_  256
#define H_   8
#define DH_  64

typedef float v2f __attribute__((ext_vector_type(2)));
typedef float v8f __attribute__((ext_vector_type(8)));

// CDNA5 fp32 matrix core: D(16x16,f32) = A(16x4,f32) x B(4x16,f32) + C
static __device__ __forceinline__ v8f wmma4(v2f a, v2f b, v8f c) {
  return __builtin_amdgcn_wmma_f32_16x16x4_f32(false, a, false, b, (short)0, c,
                                               false, false);
}

// CDNA5 async global->LDS copy (16B per lane), tracked with ASYNCcnt.
// GV addressing: 64-bit VGPR address, saddr = off.
static __device__ __forceinline__ void async_copy16(unsigned lds_off,
                                                    const void* gptr) {
  asm volatile("global_load_async_to_lds_b128 %0, %1, off"
               :
               : "v"(lds_off), "v"(gptr)
               : "memory");
}
static __device__ __forceinline__ void wait_async() {
  asm volatile("s_wait_asynccnt 0x0" ::: "memory");
}
// low 32 bits of a generic pointer to LDS = LDS byte offset
static __device__ __forceinline__ unsigned lds_off32(const void* p) {
  return (unsigned)(unsigned long long)p;
}

// ---------------------------------------------------------------------------
// Pooling: score-weighted feature pool + majority-vote label pool (exact ties)
// one block per pooled token
// ---------------------------------------------------------------------------
__global__ __launch_bounds__(256) void pool_kernel(
    const float* __restrict__ x, const float* __restrict__ scores,
    const int* __restrict__ labels, float* __restrict__ xl,
    float* __restrict__ sl, int* __restrict__ ll, int Nl, int s) {
  long gidx = blockIdx.x;  // 0..B*Nl-1
  int b = (int)(gidx / Nl);
  long t = gidx - (long)b * Nl;
  long src = (long)b * ((long)Nl * s) + t * s;
  const float* xb = x + src * DIM_;
  const float* sc = scores + src;
  float w[4];
  float wsum = 0.f, ssum = 0.f;
  for (int j = 0; j < s; ++j) {
    float v = sc[j];
    w[j] = fmaxf(v, 1e-6f);
    wsum += w[j];
    ssum += v;
  }
  for (int c = threadIdx.x; c < DIM_; c += 256) {
    float a = 0.f;
    for (int j = 0; j < s; ++j) a += xb[(long)j * DIM_ + c] * w[j];
    xl[gidx * DIM_ + c] = a / wsum;
  }
  if (threadIdx.x == 0) {
    sl[gidx] = ssum / (float)s;
    const int* lb = labels + src;
    int cnt[4] = {0, 0, 0, 0};
    float ss[4] = {0.f, 0.f, 0.f, 0.f};
    int fp[4];
    for (int c = 0; c < 4; ++c) fp[c] = s;
    for (int j = 0; j < s; ++j) {
      int L = lb[j];
      cnt[L] += 1;
      ss[L] += sc[j];
      if (fp[L] == s) fp[L] = j;
    }
    int maxc = 0;
    for (int c = 0; c < 4; ++c)
      if (cnt[c] > maxc) maxc = cnt[c];
    float s2[4];
    float maxs = -3.4e38f;
    for (int c = 0; c < 4; ++c) {
      s2[c] = (cnt[c] == maxc) ? ss[c] : -1e9f;
      maxs = fmaxf(maxs, s2[c]);
    }
    int best = 0, bestf = s + 1;
    for (int c = 0; c < 4; ++c) {
      int f = (s2[c] == maxs) ? fp[c] : s;
      if (f < bestf) {
        bestf = f;
        best = c;
      }
    }
    ll[gidx] = best;
  }
}

// ---------------------------------------------------------------------------
// Per-group focus ratio -> keep count. one block (128 threads) per (b,g)
// ---------------------------------------------------------------------------
__global__ __launch_bounds__(128) void stats_kernel(
    const int* __restrict__ labels, const float* __restrict__ scores,
    int* __restrict__ keep, int ng) {
  __shared__ int cnt[4];
  __shared__ float red[128];
  __shared__ float meansh;
  int tid = threadIdx.x;
  int bg = blockIdx.x;
  long base = (long)bg * GS_;  // b*Nl + g*128
  if (tid < 4) cnt[tid] = 0;
  __syncthreads();
  int L = labels[base + tid];
  float sc = scores[base + tid];
  atomicAdd(&cnt[L], 1);
  red[tid] = sc;
  __syncthreads();
  for (int o = 64; o > 0; o >>= 1) {
    if (tid < o) red[tid] += red[tid + o];
    __syncthreads();
  }
  if (tid == 0) meansh = red[0] / 128.f;
  __syncthreads();
  float d = sc - meansh;
  red[tid] = d * d;
  __syncthreads();
  for (int o = 64; o > 0; o >>= 1) {
    if (tid < o) red[tid] += red[tid + o];
    __syncthreads();
  }
  if (tid == 0) {
    float var = red[0] / 128.f;
    int mc = cnt[0];
    for (int c = 1; c < 4; ++c)
      if (cnt[c] > mc) mc = cnt[c];
    float purity = (float)mc / 128.f;
    float focus = 0.5f + 0.25f * purity - 0.25f * var;
    focus = fminf(fmaxf(focus, 0.25f), 0.75f);
    keep[bg] = (int)ceilf(focus * 256.f);
  }
}

// ---------------------------------------------------------------------------
// WMMA GEMM: C[M x 512] = A[M x 512] * W[512 x 512]
// block: 16 rows x 512 cols, 8 waves, each wave 16x64 via v_wmma_f32_16x16x4_f32
// A-tile staged with async global->LDS; epilogue fuses repeat(out, scale) += acc
// ---------------------------------------------------------------------------
__global__ __launch_bounds__(256) void gemm_kernel(
    const float* __restrict__ A, const float* __restrict__ W,
    float* __restrict__ C, int Nl, int repeat, int doAdd) {
  __shared__ float ldsA[16 * 516];  // padded: conflict-free column reads
  long row0 = (long)blockIdx.x * 16;
  {
    unsigned base = lds_off32(ldsA);
    const float* Ab = A + row0 * DIM_;
    for (int i = threadIdx.x; i < 2048; i += 256) {
      int fi = i << 2;
      int r = fi >> 9;
      int c = fi & 511;
      async_copy16(base + (unsigned)(r * 516 + c) * 4u, Ab + fi);
    }
  }
  wait_async();
  __syncthreads();
  int lane = threadIdx.x & 31;
  int wave = threadIdx.x >> 5;
  int half = lane >> 4, ln = lane & 15;
  int n0 = wave * 64;
  v8f acc[4] = {};
  for (int k = 0; k < DIM_; k += 4) {
    v2f a;
    a.x = ldsA[ln * 516 + k + 2 * half];
    a.y = ldsA[ln * 516 + k + 2 * half + 1];
    const float* wp = W + (long)(k + 2 * half) * DIM_ + n0 + ln;
    __builtin_prefetch(wp + 32 * DIM_, 0, 1);  // global_prefetch_b8
#pragma unroll
    for (int t = 0; t < 4; ++t) {
      v2f b;
      b.x = wp[t * 16];          // row k+2*half
      b.y = wp[DIM_ + t * 16];   // row k+2*half+1
      acc[t] = wmma4(a, b, acc[t]);
    }
  }
  int bb = (int)(row0 / Nl);  // blocks never straddle the batch boundary
  long t0 = row0 - (long)bb * Nl;
  long NlOut = (long)Nl * repeat;
#pragma unroll
  for (int t = 0; t < 4; ++t) {
#pragma unroll
    for (int r = 0; r < 8; ++r) {
      float val = acc[t][r];
      long tok = t0 + r + 8 * half;
      int col = n0 + t * 16 + ln;
      for (int rep = 0; rep < repeat; ++rep) {
        float* p = C + ((long)bb * NlOut + tok * repeat + rep) * DIM_ + col;
        if (doAdd)
          *p += val;
        else
          *p = val;
      }
    }
  }
}

// ---------------------------------------------------------------------------
// PFSA attention: one block per (b, g, h). 8 waves, dynamic LDS:
//   ldsQ  128x68, ldsKV 256x68 (k then reused for v), ldsS 128x260 (sim/attn)
// q/k/v windows staged with async global->LDS (ASYNCcnt), sim/out via WMMA
// ---------------------------------------------------------------------------
__global__ __launch_bounds__(256) void attn_kernel(
    const float* __restrict__ q, const float* __restrict__ k,
    const float* __restrict__ v, const int* __restrict__ keep,
    const float* __restrict__ betaPtr, const float* __restrict__ prior,
    float* __restrict__ attnSave, float* __restrict__ outbuf, int Nl,
    int ngPrev, int saveFlag) {
  extern __shared__ float smem[];
  float* ldsQ = smem;                          // 128*68
  float* ldsKV = smem + 128 * 68;              // 256*68
  float* ldsS = smem + 128 * 68 + 256 * 68;    // 128*260
  const int ng = Nl / GS_;
  int blk = blockIdx.x;
  int h = blk & 7;
  int g = (blk >> 3) % ng;
  int b = blk / (8 * ng);
  unsigned qoff = lds_off32(ldsQ);
  unsigned kvoff = lds_off32(ldsKV);
  long qbase = ((long)b * Nl + (long)g * GS_) * DIM_ + h * DH_;
  for (int i = threadIdx.x; i < 128 * 16; i += 256) {
    int rq = i >> 4;
    int c = (i & 15) << 2;
    async_copy16(qoff + (unsigned)(rq * 68 + c) * 4u,
                 q + qbase + (long)rq * DIM_ + c);
  }
  for (int i = threadIdx.x; i < 256 * 16; i += 256) {
    int w = i >> 4;
    int c = (i & 15) << 2;
    int p = g * GS_ + w;
    int row = (p < Nl) ? p : (2 * Nl - 1 - p);  // reflected tail window
    async_copy16(kvoff + (unsigned)(w * 68 + c) * 4u,
                 k + ((long)b * Nl + row) * DIM_ + h * DH_ + c);
  }
  wait_async();
  __syncthreads();
  int lane = threadIdx.x & 31;
  int wave = threadIdx.x >> 5;
  int half = lane >> 4, ln = lane & 15;
  int qr0 = wave * 16;
  {  // sim = q k^T / 8 via WMMA -> ldsS
    v8f sim[16] = {};
    for (int kd = 0; kd < DH_; kd += 4) {
      v2f a;
      a.x = ldsQ[(qr0 + ln) * 68 + kd + 2 * half];
      a.y = ldsQ[(qr0 + ln) * 68 + kd + 2 * half + 1];
#pragma unroll
      for (int t = 0; t < 16; ++t) {
        v2f bb;
        bb.x = ldsKV[(t * 16 + ln) * 68 + kd + 2 * half];
        bb.y = ldsKV[(t * 16 + ln) * 68 + kd + 2 * half + 1];
        sim[t] = wmma4(a, bb, sim[t]);
      }
    }
#pragma unroll
    for (int t = 0; t < 16; ++t) {
#pragma unroll
      for (int r = 0; r < 8; ++r)
        ldsS[(qr0 + r + 8 * half) * 260 + t * 16 + ln] = sim[t][r] * 0.125f;
    }
  }
  __syncthreads();
  {  // exact top-k (stable double-argsort rank), softmax, prior blend
    int rowm = threadIdx.x >> 1;
    int sub = threadIdx.x & 1;
    float* srow = ldsS + rowm * 260;
    int kp = keep[b * ng + g];
    unsigned mbits[4] = {0, 0, 0, 0};
    for (int kk = 0; kk < 128; ++kk) {
      int key = sub * 128 + kk;
      float val = srow[key];
      int cnt = 0;
      for (int j = 0; j < 256; ++j) {
        float vj = srow[j];
        cnt += ((vj > val) || ((vj == val) && (j < key))) ? 1 : 0;
      }
      if (cnt < kp) mbits[kk >> 5] |= (1u << (kk & 31));
    }
    __syncthreads();
    float beta = 0.f, wlerp = 0.f;
    const float* prowLo = nullptr;
    const float* prowHi = nullptr;
    if (ngPrev > 0) {
      beta = 1.f / (1.f + expf(-betaPtr[0]));
      float posf = ((float)g + 0.5f) * ((float)ngPrev / (float)ng) - 0.5f;
      float fl = floorf(posf);
      int lo = (int)fl;
      lo = lo < 0 ? 0 : (lo > ngPrev - 1 ? ngPrev - 1 : lo);
      int hi = lo + 1;
      hi = hi > ngPrev - 1 ? ngPrev - 1 : hi;
      wlerp = fminf(fmaxf(posf - fl, 0.f), 1.f);
      prowLo = prior + ((((long)b * ngPrev + lo) * H_ + h) * GS_ + rowm) * WS_;
      prowHi = prior + ((((long)b * ngPrev + hi) * H_ + h) * GS_ + rowm) * WS_;
    }
    float rmax = -3.4e38f;
    for (int kk = 0; kk < 128; ++kk)
      if (mbits[kk >> 5] & (1u << (kk & 31)))
        rmax = fmaxf(rmax, srow[sub * 128 + kk]);
    rmax = fmaxf(rmax, __shfl_xor(rmax, 1));
    float esum = 0.f, psum = 0.f;
    for (int kk = 0; kk < 128; ++kk) {
      int key = sub * 128 + kk;
      if (mbits[kk >> 5] & (1u << (kk & 31))) esum += expf(srow[key] - rmax);
      if (ngPrev > 0) {
        float pv = (1.f - wlerp) * prowLo[key] + wlerp * prowHi[key];
        psum += fmaxf(pv, 0.f);
      }
    }
    esum += __shfl_xor(esum, 1);
    psum += __shfl_xor(psum, 1);
    float pnorm = (ngPrev > 0) ? (beta / (psum + 1e-9f)) : 0.f;
    float* savRow = nullptr;
    if (saveFlag)
      savRow = attnSave + ((((long)b * ng + g) * H_ + h) * GS_ + rowm) * WS_;
    for (int kk = 0; kk < 128; ++kk) {
      int key = sub * 128 + kk;
      float a = 0.f;
      if (mbits[kk >> 5] & (1u << (kk & 31)))
        a = expf(srow[key] - rmax) / esum;
      if (ngPrev > 0) {
        float pv = fmaxf((1.f - wlerp) * prowLo[key] + wlerp * prowHi[key], 0.f);
        a = (1.f - beta) * a + pv * pnorm;
      }
      srow[key] = a;
      if (saveFlag) savRow[key] = a;
    }
  }
  __syncthreads();
  for (int i = threadIdx.x; i < 256 * 16; i += 256) {  // v overwrites k region
    int w = i >> 4;
    int c = (i & 15) << 2;
    int p = g * GS_ + w;
    int row = (p < Nl) ? p : (2 * Nl - 1 - p);
    async_copy16(kvoff + (unsigned)(w * 68 + c) * 4u,
                 v + ((long)b * Nl + row) * DIM_ + h * DH_ + c);
  }
  wait_async();
  __syncthreads();
  {  // out = attn @ v via WMMA
    v8f o[4] = {};
    for (int kk = 0; kk < WS_; kk += 4) {
      v2f a;
      a.x = ldsS[(qr0 + ln) * 260 + kk + 2 * half];
      a.y = ldsS[(qr0 + ln) * 260 + kk + 2 * half + 1];
#pragma unroll
      for (int t = 0; t < 4; ++t) {
        v2f bb;
        bb.x = ldsKV[(kk + 2 * half) * 68 + t * 16 + ln];
        bb.y = ldsKV[(kk + 2 * half + 1) * 68 + t * 16 + ln];
        o[t] = wmma4(a, bb, o[t]);
      }
    }
#pragma unroll
    for (int t = 0; t < 4; ++t) {
#pragma unroll
      for (int r = 0; r < 8; ++r)
        outbuf[((long)b * Nl + g * GS_ + qr0 + r + 8 * half) * DIM_ + h * DH_ +
               t * 16 + ln] = o[t][r];
    }
  }
}

// ---------------------------------------------------------------------------
// restore_sorted_tokens: out[b, idx[b,j], :] = acc[b, j, :]
// ---------------------------------------------------------------------------
__global__ __launch_bounds__(256) void scatter_kernel(
    const float* __restrict__ acc, const int* __restrict__ idx,
    float* __restrict__ out) {
  long i = (long)blockIdx.x * 256 + threadIdx.x;  // over B*N*128 float4s
  long rowc = i >> 7;
  int c4 = (int)(i & 127);
  int b = (int)(rowc >> 13);
  int dest = idx[rowc];
  float4 val = ((const float4*)acc)[rowc * 128 + c4];
  ((float4*)out)[(((long)b << 13) + dest) * 128 + c4] = val;
}

// ---------------------------------------------------------------------------
// Workspace layout (bytes)
// ---------------------------------------------------------------------------
static constexpr size_t OFF_ACC = 0;
static constexpr size_t OFF_XL = OFF_ACC + (size_t)B_ * N_ * DIM_ * 4;
static constexpr size_t OFF_Q = OFF_XL + (size_t)B_ * (N_ / 2) * DIM_ * 4;
static constexpr size_t OFF_K = OFF_Q + (size_t)B_ * N_ * DIM_ * 4;
static constexpr size_t OFF_V = OFF_K + (size_t)B_ * N_ * DIM_ * 4;
static constexpr size_t OFF_AO = OFF_V + (size_t)B_ * N_ * DIM_ * 4;
static constexpr size_t OFF_ATA = OFF_AO + (size_t)B_ * N_ * DIM_ * 4;
static constexpr size_t OFF_ATB = OFF_ATA + (size_t)B_ * 16 * H_ * GS_ * WS_ * 4;
static constexpr size_t OFF_SL = OFF_ATB + (size_t)B_ * 32 * H_ * GS_ * WS_ * 4;
static constexpr size_t OFF_LL = OFF_SL + (size_t)B_ * (N_ / 2) * 4;
static constexpr size_t OFF_KP = OFF_LL + (size_t)B_ * (N_ / 2) * 4;
static constexpr size_t WS_NEED = OFF_KP + 1024;

extern "C" void kernel_launch(void* const* d_in, const int* in_sizes, int n_in,
                              void* d_out, int out_size, void* d_ws,
                              size_t ws_size, hipStream_t stream) {
  (void)in_sizes; (void)n_in; (void)out_size;
  if (ws_size < WS_NEED) return;
  const float* x = (const float*)d_in[0];
  const int* labels = (const int*)d_in[1];
  const float* scores = (const float*)d_in[2];
  const int* idx_last = (const int*)d_in[3];
  const float* Wq = (const float*)d_in[4];
  const float* Wk = (const float*)d_in[5];
  const float* Wv = (const float*)d_in[6];
  const float* Wp = (const float*)d_in[7];
  const float* beta = (const float*)d_in[8];
  float* out = (float*)d_out;

  char* ws = (char*)d_ws;
  float* acc = (float*)(ws + OFF_ACC);
  float* xl = (float*)(ws + OFF_XL);
  float* qb = (float*)(ws + OFF_Q);
  float* kb = (float*)(ws + OFF_K);
  float* vb = (float*)(ws + OFF_V);
  float* ao = (float*)(ws + OFF_AO);
  float* atA = (float*)(ws + OFF_ATA);
  float* atB = (float*)(ws + OFF_ATB);
  float* sl = (float*)(ws + OFF_SL);
  int* ll = (int*)(ws + OFF_LL);
  int* kp = (int*)(ws + OFF_KP);

  const int scales[3] = {4, 2, 1};
  const size_t ATTN_SHMEM = (128 * 68 + 256 * 68 + 128 * 260) * sizeof(float);

  for (int lvl = 0; lvl < 3; ++lvl) {
    int s = scales[lvl];
    int Nl = N_ / s;
    int ng = Nl / GS_;
    const float* xcur;
    const float* scur;
    const int* lcur;
    if (s > 1) {
      pool_kernel<<<B_ * Nl, 256, 0, stream>>>(x, scores, labels, xl, sl, ll,
                                               Nl, s);
      xcur = xl; scur = sl; lcur = ll;
    } else {
      xcur = x; scur = scores; lcur = labels;
    }
    stats_kernel<<<B_ * ng, 128, 0, stream>>>(lcur, scur, kp, ng);
    int mblocks = (B_ * Nl) / 16;
    gemm_kernel<<<mblocks, 256, 0, stream>>>(
        xcur, Wq + (size_t)lvl * DIM_ * DIM_, qb, Nl, 1, 0);
    gemm_kernel<<<mblocks, 256, 0, stream>>>(
        xcur, Wk + (size_t)lvl * DIM_ * DIM_, kb, Nl, 1, 0);
    gemm_kernel<<<mblocks, 256, 0, stream>>>(
        xcur, Wv + (size_t)lvl * DIM_ * DIM_, vb, Nl, 1, 0);
    const float* prior = (lvl == 0) ? nullptr : (lvl == 1 ? atA : atB);
    int ngPrev = (lvl == 0) ? 0 : (lvl == 1 ? 16 : 32);
    float* save = (lvl == 0) ? atA : atB;
    int saveFlag = (lvl < 2) ? 1 : 0;
    attn_kernel<<<B_ * ng * H_, 256, ATTN_SHMEM, stream>>>(
        qb, kb, vb, kp, beta + lvl, prior, save, ao, Nl, ngPrev, saveFlag);
    gemm_kernel<<<mblocks, 256, 0, stream>>>(
        ao, Wp + (size_t)lvl * DIM_ * DIM_, acc, Nl, s, lvl > 0 ? 1 : 0);
  }
  scatter_kernel<<<(int)(((long)B_ * N_ * 128) / 256), 256, 0, stream>>>(
      acc, idx_last, out);
}